// ConvexSampler_39006892982339
// MI455X (gfx1250) — compile-verified
//
#include <hip/hip_runtime.h>
#include <stdint.h>

// Problem geometry (from the reference)
#define BATCH      8192
#define FEAT       768
#define F4ROW      192               // 768 floats / 4 per float4
#define NCONVEX    32768
#define NROWS_OUT  (BATCH + NCONVEX) // 40960
#define UNSEEN_F   150.0f
#define ROWS_PER_BLOCK 16

typedef __attribute__((ext_vector_type(4))) float v4f;
typedef int v4i_ __attribute__((vector_size(4 * sizeof(int))));  // builtin's pointee type

// ---------------------------------------------------------------------------
// CDNA5 async global->LDS staging (ASYNCcnt path). Builtin if available,
// otherwise raw gfx1250 inline asm (portable across both toolchains).
// ---------------------------------------------------------------------------
__device__ __forceinline__ void async_g2l_b128(const v4f* gsrc, v4f* lds_dst) {
#if __has_builtin(__builtin_amdgcn_global_load_async_to_lds_b128)
  __builtin_amdgcn_global_load_async_to_lds_b128(
      (__attribute__((address_space(1))) v4i_*)
          (__attribute__((address_space(1))) void*)gsrc,
      (__attribute__((address_space(3))) v4i_*)
          (__attribute__((address_space(3))) void*)lds_dst,
      0, 0);
#else
  uint32_t lofs =
      (uint32_t)(uintptr_t)(__attribute__((address_space(3))) void*)lds_dst;
  asm volatile("global_load_async_to_lds_b128 %0, %1, off"
               :: "v"(lofs), "v"((uint64_t)(uintptr_t)gsrc)
               : "memory");
#endif
}

template <int N>
__device__ __forceinline__ void wait_asynccnt() {
#if __has_builtin(__builtin_amdgcn_s_wait_asynccnt)
  __builtin_amdgcn_s_wait_asynccnt(N);
#else
  asm volatile("s_wait_asynccnt %0" :: "i"(N) : "memory");
#endif
}

// ---------------------------------------------------------------------------
// Kernel 1: convex rows. One block = 192 threads (6 waves) = one 768-float
// row per stage. Double-buffered async staging of the two gathered source
// rows into LDS; each lane reads back exactly the 16B it staged (no barriers
// needed). Output stored non-temporally so the 96MB stream does not evict z
// (24MB, re-read ~8x by the gather) from the 192MB L2.
// ---------------------------------------------------------------------------
__global__ __launch_bounds__(F4ROW) void convex_rows_kernel(
    const float* __restrict__ z,
    const int*   __restrict__ idx_i,
    const int*   __restrict__ idx_j,
    const float* __restrict__ s,
    float*       __restrict__ out) {
  __shared__ v4f bufA[2][F4ROW];   // idx_i source row, 2-deep pipeline
  __shared__ v4f bufB[2][F4ROW];   // idx_j source row

  const v4f* z4   = (const v4f*)z;
  v4f*       out4 = (v4f*)out;
  const int  tid  = threadIdx.x;
  const int  r0   = blockIdx.x * ROWS_PER_BLOCK;

  // Prologue: stage row r0 into slot 0 (2 async b128 ops per lane).
  {
    const int i = idx_i[r0];
    const int j = idx_j[r0];
    async_g2l_b128(z4 + (size_t)i * F4ROW + tid, &bufA[0][tid]);
    async_g2l_b128(z4 + (size_t)j * F4ROW + tid, &bufB[0][tid]);
  }

  for (int k = 0; k < ROWS_PER_BLOCK; ++k) {
    const int sl = k & 1;

    // Stage row k+1 into the other slot, then wait for row k's 2 ops
    // (async loads complete in order: <=2 outstanding => stage k done).
    if (k + 1 < ROWS_PER_BLOCK) {
      const int rn = r0 + k + 1;
      const int i  = idx_i[rn];      // uniform -> scalar loads
      const int j  = idx_j[rn];
      async_g2l_b128(z4 + (size_t)i * F4ROW + tid, &bufA[sl ^ 1][tid]);
      async_g2l_b128(z4 + (size_t)j * F4ROW + tid, &bufB[sl ^ 1][tid]);
      wait_asynccnt<2>();
    } else {
      wait_asynccnt<0>();
    }

    const int   r  = r0 + k;
    const float sv = s[r];
    const float tv = 1.0f - sv;

    const v4f a = bufA[sl][tid];
    const v4f b = bufB[sl][tid];
    v4f o;
    o.x = sv * a.x + tv * b.x;
    o.y = sv * a.y + tv * b.y;
    o.z = sv * a.z + tv * b.z;
    o.w = sv * a.w + tv * b.w;

    __builtin_nontemporal_store(o, out4 + (size_t)(BATCH + r) * F4ROW + tid);
  }
}

// ---------------------------------------------------------------------------
// Kernel 2: straight copy of z into the output prefix. Regular-temporal
// loads (prime z into L2 for the gather kernel), non-temporal stores.
// ---------------------------------------------------------------------------
__global__ __launch_bounds__(256) void copy_z_kernel(
    const float* __restrict__ z, float* __restrict__ out) {
  const size_t g   = (size_t)blockIdx.x * 256 + threadIdx.x;
  const v4f*   z4  = (const v4f*)z;
  v4f*         o4  = (v4f*)out;
  v4f v = z4[g];
  __builtin_nontemporal_store(v, o4 + g);
}

// ---------------------------------------------------------------------------
// Kernel 3: labels. Output buffer dtype is float, so write values.
// ---------------------------------------------------------------------------
__global__ __launch_bounds__(256) void labels_kernel(
    const int* __restrict__ labels, float* __restrict__ out_lab) {
  const int g = blockIdx.x * 256 + threadIdx.x;
  if (g < NROWS_OUT) {
    const float v = (g < BATCH) ? (float)labels[g] : UNSEEN_F;
    __builtin_nontemporal_store(v, out_lab + g);
  }
}

// ---------------------------------------------------------------------------
extern "C" void kernel_launch(void* const* d_in, const int* in_sizes, int n_in,
                              void* d_out, int out_size, void* d_ws,
                              size_t ws_size, hipStream_t stream) {
  (void)in_sizes; (void)n_in; (void)out_size; (void)d_ws; (void)ws_size;

  const float* z      = (const float*)d_in[0];
  const int*   labels = (const int*)  d_in[1];
  const int*   idx_i  = (const int*)  d_in[2];
  const int*   idx_j  = (const int*)  d_in[3];
  const float* s      = (const float*)d_in[4];

  float* out     = (float*)d_out;
  float* out_lab = out + (size_t)NROWS_OUT * FEAT;

  // Copy first: primes z into L2 before the gather kernel re-reads it ~8x.
  copy_z_kernel<<<(BATCH * F4ROW) / 256, 256, 0, stream>>>(z, out);
  convex_rows_kernel<<<NCONVEX / ROWS_PER_BLOCK, F4ROW, 0, stream>>>(
      z, idx_i, idx_j, s, out);
  labels_kernel<<<(NROWS_OUT + 255) / 256, 256, 0, stream>>>(labels, out_lab);
}